// Imbalanced_SinkhornKnopp_72060961292676
// MI455X (gfx1250) — compile-verified
//
#include <hip/hip_runtime.h>
#include <math.h>

#define QUEUE   32768
#define NCLUS   1000
#define NTOT    (QUEUE * NCLUS)
#define NCHUNK  64            // queue chunks for the WMMA rowsum (deterministic partials)
#define CPAD    1024          // padded cluster stride in scratch

typedef __attribute__((ext_vector_type(2))) float v2f;
typedef __attribute__((ext_vector_type(8))) float v8f;

// ---------------------------------------------------------------- global max
__global__ __launch_bounds__(256) void k_max_partial(const float* __restrict__ ml,
                                                     float* __restrict__ part, int n) {
    __shared__ float sm[256];
    float m = -INFINITY;
    for (int i = blockIdx.x * blockDim.x + threadIdx.x; i < n; i += gridDim.x * blockDim.x)
        m = fmaxf(m, ml[i]);
    sm[threadIdx.x] = m;
    __syncthreads();
    for (int s = 128; s > 0; s >>= 1) {
        if (threadIdx.x < s) sm[threadIdx.x] = fmaxf(sm[threadIdx.x], sm[threadIdx.x + s]);
        __syncthreads();
    }
    if (threadIdx.x == 0) part[blockIdx.x] = sm[0];
}

// scal[0]=global max, scal[1]=log(sigmoid(k)), scal[2]=softmax denominator of K
__global__ __launch_bounds__(256) void k_finalize(const float* __restrict__ part, int nparts,
                                                  const float* __restrict__ learn_k,
                                                  float* __restrict__ scal) {
    __shared__ float sm[256];
    float m = -INFINITY;
    for (int i = threadIdx.x; i < nparts; i += 256) m = fmaxf(m, part[i]);
    sm[threadIdx.x] = m;
    __syncthreads();
    for (int s = 128; s > 0; s >>= 1) {
        if (threadIdx.x < s) sm[threadIdx.x] = fmaxf(sm[threadIdx.x], sm[threadIdx.x + s]);
        __syncthreads();
    }
    float maxv = sm[0];
    __syncthreads();
    float k = learn_k[0];
    float L = -log1pf(expf(-k));            // log(sigmoid(k)), stable for k>0
    float p = 0.f;
    for (int i = threadIdx.x; i < NCLUS; i += 256) {
        float w = (float)i / (float)(NCLUS - 1);
        p += expf(w * L);
    }
    sm[threadIdx.x] = p;
    __syncthreads();
    for (int s = 128; s > 0; s >>= 1) {
        if (threadIdx.x < s) sm[threadIdx.x] += sm[threadIdx.x + s];
        __syncthreads();
    }
    if (threadIdx.x == 0) { scal[0] = maxv; scal[1] = L; scal[2] = sm[0]; }
}

// ------------------------------------------------------- Q0 = exp((ml-max)/eps)
__global__ __launch_bounds__(256) void k_exp(const float* __restrict__ ml,
                                             float* __restrict__ q,
                                             const float* __restrict__ scal, int n4) {
    const float maxv = scal[0];
    const float c = 2.0f * 1.4426950408889634f;   // (1/eps)*log2(e)
    const float4* in = (const float4*)ml;
    float4* out = (float4*)q;
    for (int i = blockIdx.x * blockDim.x + threadIdx.x; i < n4; i += gridDim.x * blockDim.x) {
        float4 v = in[i], r;
        r.x = exp2f((v.x - maxv) * c);
        r.y = exp2f((v.y - maxv) * c);
        r.z = exp2f((v.z - maxv) * c);
        r.w = exp2f((v.w - maxv) * c);
        out[i] = r;
    }
}

// ---------------------------------------------- rowsum over n via WMMA f32 16x16x4
// A tile = Q^T[16 clusters x 4 queue rows], B = ones(4x16): D accumulates per-cluster
// sums (replicated over N). One wave per block; uniform flow around WMMA (EXEC all-1).
// Each block writes its 16 partial sums (no atomics) into partial[chunk][c] slot;
// a deterministic reduce kernel folds the NCHUNK partials in fixed order.
__global__ __launch_bounds__(32) void k_wmma_rowsum(const float* __restrict__ q,
                                                    float* __restrict__ partial) {
    const int lane = threadIdx.x;
    const int m    = lane & 15;
    const int kh   = lane >> 4;                 // K-half: 0 -> K=0,1 ; 1 -> K=2,3
    const int c0   = blockIdx.x * 16;
    const int c    = c0 + m;
    const float msk = (c < NCLUS) ? 1.0f : 0.0f;
    const int cl   = (c < NCLUS) ? c : (NCLUS - 1);
    const int nper = QUEUE / NCHUNK;
    const int n0   = blockIdx.y * nper;

    v8f acc = {0.f, 0.f, 0.f, 0.f, 0.f, 0.f, 0.f, 0.f};
    v2f b;  b.x = 1.0f; b.y = 1.0f;             // ones vector (all K, all N)

#pragma unroll 4
    for (int nn = n0; nn < n0 + nper; nn += 4) {
        const int r0 = nn + 2 * kh;
        v2f a;
        a.x = q[(size_t)r0 * NCLUS + cl] * msk;        // K = 2*kh
        a.y = q[(size_t)(r0 + 1) * NCLUS + cl] * msk;  // K = 2*kh + 1
        acc = __builtin_amdgcn_wmma_f32_16x16x4_f32(false, a, false, b,
                                                    (short)0, acc, false, false);
    }
    // D layout: elem i of lane L -> M = i + 8*(L>=16), N = L%16. Take the N=0 column:
    // lane 0 holds M=0..7, lane 16 holds M=8..15 -> two contiguous float4 stores each.
    if (lane == 0 || lane == 16) {
        float* dst = partial + (size_t)blockIdx.y * CPAD + c0 + (lane >> 4) * 8;
        float4 lo, hi;
        lo.x = acc[0]; lo.y = acc[1]; lo.z = acc[2]; lo.w = acc[3];
        hi.x = acc[4]; hi.y = acc[5]; hi.z = acc[6]; hi.w = acc[7];
        *(float4*)(dst)     = lo;
        *(float4*)(dst + 4) = hi;
    }
}

// deterministic fold of the NCHUNK partial rowsums (fixed order)
__global__ void k_reduce_rs(const float* __restrict__ partial, float* __restrict__ rs) {
    int c = blockIdx.x * blockDim.x + threadIdx.x;
    if (c < NCLUS) {
        float s = 0.f;
#pragma unroll 8
        for (int j = 0; j < NCHUNK; ++j) s += partial[(size_t)j * CPAD + c];
        rs[c] = s;
    }
}

// -------------------------- rank from R0 (stable desc), Keff = K[rank], iter-1 scale
__global__ __launch_bounds__(1024) void k_rank(const float* __restrict__ rs,
                                               const float* __restrict__ scal,
                                               float* __restrict__ keff,
                                               float* __restrict__ scale) {
    __shared__ float sR[NCLUS];
    for (int i = threadIdx.x; i < NCLUS; i += blockDim.x) sR[i] = rs[i];
    __syncthreads();
    const float L = scal[1], kden = scal[2];
    for (int c = threadIdx.x; c < NCLUS; c += blockDim.x) {
        const float v = sR[c];
        int rank = 0;
        for (int j = 0; j < NCLUS; ++j) {
            const float u = sR[j];
            rank += (u > v) || (u == v && j < c);
        }
        const float w  = (float)rank / (float)(NCLUS - 1);
        const float ke = expf(w * L) / kden;
        keff[c]  = ke;
        scale[c] = ke / v;
    }
}

__global__ void k_scale(const float* __restrict__ rs, const float* __restrict__ keff,
                        float* __restrict__ scale) {
    int c = blockIdx.x * blockDim.x + threadIdx.x;
    if (c < NCLUS) scale[c] = keff[c] / rs[c];
}

// ------------- fused Sinkhorn step: row-scale + clip + colsum + col-normalize.
// One wave per queue row n (1000 contiguous f32 = 250 float4; slot 7 only lanes<26).
__global__ __launch_bounds__(256) void k_iter(float* __restrict__ q,
                                              const float* __restrict__ scale,
                                              int finalIter) {
    const int lane = threadIdx.x & 31;
    const int wid  = (blockIdx.x * blockDim.x + threadIdx.x) >> 5;
    const int nw   = (gridDim.x * blockDim.x) >> 5;

    float4 s4[8];
    const float4* sc4 = (const float4*)scale;
#pragma unroll
    for (int kb = 0; kb < 8; ++kb)
        if (kb < 7 || lane < 26) s4[kb] = sc4[kb * 32 + lane];

    for (int n = wid; n < QUEUE; n += nw) {
        float4* row = (float4*)(q + (size_t)n * NCLUS);
        float4 v[8];
        float part = 0.f;
#pragma unroll
        for (int kb = 0; kb < 8; ++kb) {
            if (kb < 7 || lane < 26) {
                float4 t = row[kb * 32 + lane];
                t.x = fmaxf(t.x * s4[kb].x, 1e-16f);
                t.y = fmaxf(t.y * s4[kb].y, 1e-16f);
                t.z = fmaxf(t.z * s4[kb].z, 1e-16f);
                t.w = fmaxf(t.w * s4[kb].w, 1e-16f);
                v[kb] = t;
                part += t.x + t.y + t.z + t.w;
            }
        }
#pragma unroll
        for (int off = 16; off > 0; off >>= 1)
            part += __shfl_xor(part, off, 32);
        const float inv = finalIter ? (1.0f / part)                    // *B cancels /B
                                    : (1.0f / (part * (float)QUEUE));
#pragma unroll
        for (int kb = 0; kb < 8; ++kb) {
            if (kb < 7 || lane < 26) {
                float4 t = v[kb];
                t.x *= inv; t.y *= inv; t.z *= inv; t.w *= inv;
                row[kb * 32 + lane] = t;
            }
        }
    }
}

extern "C" void kernel_launch(void* const* d_in, const int* in_sizes, int n_in,
                              void* d_out, int out_size, void* d_ws, size_t ws_size,
                              hipStream_t stream) {
    (void)in_sizes; (void)n_in; (void)out_size; (void)ws_size;
    const float* ml = (const float*)d_in[0];
    const float* lk = (const float*)d_in[1];
    float* q   = (float*)d_out;          // working matrix in output layout [n, c] = Q^T
    float* ws  = (float*)d_ws;
    float* part    = ws;                 // 1024 floats (512 used)
    float* rs      = ws + 1024;          // row sums R[c]
    float* keff    = ws + 2048;          // K[rank(c)]
    float* scale   = ws + 3072;          // Keff[c] / R[c]
    float* scal    = ws + 4096;          // [max, log sigmoid(k), K denom]
    float* partial = ws + 8192;          // NCHUNK x CPAD rowsum partials (256 KB)

    k_max_partial<<<512, 256, 0, stream>>>(ml, part, NTOT);
    k_finalize<<<1, 256, 0, stream>>>(part, 512, lk, scal);
    k_exp<<<2048, 256, 0, stream>>>(ml, q, scal, NTOT / 4);

    const dim3 gw((NCLUS + 15) / 16, NCHUNK);   // 63 cluster tiles x 64 queue chunks
    // iteration 1 (global /sum cancels in the row-normalize)
    k_wmma_rowsum<<<gw, 32, 0, stream>>>(q, partial);
    k_reduce_rs<<<4, 256, 0, stream>>>(partial, rs);
    k_rank<<<1, 1024, 0, stream>>>(rs, scal, keff, scale);
    k_iter<<<512, 256, 0, stream>>>(q, scale, 0);
    // iteration 2
    k_wmma_rowsum<<<gw, 32, 0, stream>>>(q, partial);
    k_reduce_rs<<<4, 256, 0, stream>>>(partial, rs);
    k_scale<<<4, 256, 0, stream>>>(rs, keff, scale);
    k_iter<<<512, 256, 0, stream>>>(q, scale, 0);
    // iteration 3 (final: *B cancels /B)
    k_wmma_rowsum<<<gw, 32, 0, stream>>>(q, partial);
    k_reduce_rs<<<4, 256, 0, stream>>>(partial, rs);
    k_scale<<<4, 256, 0, stream>>>(rs, keff, scale);
    k_iter<<<512, 256, 0, stream>>>(q, scale, 1);
}